// Qubits2Model_17935783428347
// MI455X (gfx1250) — compile-verified
//
#include <hip/hip_runtime.h>

// CDNA5 / gfx1250, wave32. One workgroup, one wave: latency-bound sequential
// chain of 99 tiny unitaries. Both half-chains run simultaneously as 8x8
// real-ified blocks packed block-diagonally into one 16x16 f32 tile, advanced
// with V_WMMA_F32_16X16X4_F32. All per-step trig is precomputed in parallel
// across the 32 lanes (fp32 hardware sin/cos) and staged in LDS, so the
// sequential critical path is only LDS traffic + WMMA.

typedef float v2f __attribute__((ext_vector_type(2)));
typedef float v8f __attribute__((ext_vector_type(8)));

#define N_STEPS 99
#define HALF_CHAIN 50      // lo chain: M_0..M_49 ; hi chain: M_50..M_98 (+ identity pad)
#define ROWP 17            // padded LDS row stride (floats) -> bank-conflict-free

// Entry (r,c) of the real-ified 8x8 step unitary  [[X,-Y],[Y,X]]
// for the block-diagonal complex 4x4:
//   block01: [[ca,  sa*sphi - i*sa*cphi], [-sa*sphi - i*sa*cphi, ca]]
//   block23: same with (cb, sb).
__device__ __forceinline__ float mentry(int r, int c,
                                        float ca, float sa, float cb, float sb,
                                        float sphi, float cphi) {
  int rr = r & 3, cc = c & 3;
  int ri = r >> 2, ci = c >> 2;      // 0 = real-part band, 1 = imag-part band
  float x = 0.f, y = 0.f;
  if (rr == cc)                 x = (rr < 2) ? ca : cb;
  else if (rr == 0 && cc == 1)  x =  sa * sphi;
  else if (rr == 1 && cc == 0)  x = -sa * sphi;
  else if (rr == 2 && cc == 3)  x =  sb * sphi;
  else if (rr == 3 && cc == 2)  x = -sb * sphi;
  if ((rr == 0 && cc == 1) || (rr == 1 && cc == 0)) y = -sa * cphi;
  if ((rr == 2 && cc == 3) || (rr == 3 && cc == 2)) y = -sb * cphi;
  if (ri == ci) return x;
  return ri ? y : -y;                // bottom-left = +Y, top-right = -Y
}

// Write one 8x8 real-ified step matrix into an LDS 16x16 tile at (br,bc).
__device__ __forceinline__ void write_block(float* tile, int br, int bc,
                                            float ca, float sa, float cb, float sb,
                                            float sphi, float cphi, int lane) {
  #pragma unroll
  for (int e0 = 0; e0 < 64; e0 += 32) {            // uniform trip count: EXEC stays full
    int e = e0 + lane;
    int r = e >> 3, c = e & 7;
    tile[(br + r) * ROWP + (bc + c)] = mentry(r, c, ca, sa, cb, sb, sphi, cphi);
  }
}

__global__ __launch_bounds__(32)
void qubits2_bell_kernel(const double* __restrict__ theta,
                         const double* __restrict__ phi,
                         const double* __restrict__ psi,   // complex128[4] = interleaved re,im
                         const double* __restrict__ Tptr,
                         float* __restrict__ out) {
  __shared__ float TA[16 * ROWP];      // step tile  diag(M_k, M_{50+k})
  __shared__ float P [16 * ROWP];      // running    diag(P_lo, P_hi)
  __shared__ float SC[2 * N_STEPS];    // precomputed {sin(phi_k), cos(phi_k)}

  const int lane = threadIdx.x;     // wave32: 0..31
  const int half = lane >> 4;       // K-half selector for A/B operands
  const int mn   = lane & 15;       // M (for A) / N (for B, C, D)

  const double T  = Tptr[0];
  const double dt = T / (double)N_STEPS;
  const float ca = (float)cos(0.5 * dt);
  const float sa = (float)sin(0.5 * dt);
  const float cb = (float)cos(dt * 0.70710678118654752440);
  const float sb = (float)sin(dt * 0.70710678118654752440);

  // Phase 0: all 99 sin/cos(phi_k) computed in parallel across the lanes
  // (fp32 hardware transcendentals; phi in [0,2pi) so no range issues).
  for (int k = lane; k < N_STEPS; k += 32) {
    float s, c;
    __sincosf((float)phi[k], &s, &c);
    SC[2 * k]     = s;
    SC[2 * k + 1] = c;
  }

  // Zero tiles (off-diagonal 8x8 quadrants stay zero forever).
  for (int i = lane; i < 16 * ROWP; i += 32) { TA[i] = 0.f; P[i] = 0.f; }
  __syncthreads();

  // Init running tile P = diag(M_0, M_50).
  write_block(P, 0, 0, ca, sa, cb, sb, SC[0], SC[1], lane);
  write_block(P, 8, 8, ca, sa, cb, sb, SC[2 * HALF_CHAIN], SC[2 * HALF_CHAIN + 1], lane);
  __syncthreads();

  // 49 steps: P <- diag(M_k, M_{50+k}) x P   (left-multiplied chain, both halves at once)
  for (int k = 1; k < HALF_CHAIN; ++k) {
    write_block(TA, 0, 0, ca, sa, cb, sb, SC[2 * k], SC[2 * k + 1], lane);
    int hi = HALF_CHAIN + k;
    if (hi < N_STEPS) {
      write_block(TA, 8, 8, ca, sa, cb, sb, SC[2 * hi], SC[2 * hi + 1], lane);
    } else {
      write_block(TA, 8, 8, 1.f, 0.f, 1.f, 0.f, 0.f, 0.f, lane);   // identity pad (M_99)
    }
    __syncthreads();

    // Load A (16x4 K-chunks) and B (4x16 K-chunks) operands from LDS.
    v2f a[4], b[4];
    #pragma unroll
    for (int q = 0; q < 4; ++q) {
      int kk = 4 * q + 2 * half;
      a[q].x = TA[mn * ROWP + kk];
      a[q].y = TA[mn * ROWP + kk + 1];
      b[q].x = P[kk * ROWP + mn];
      b[q].y = P[(kk + 1) * ROWP + mn];
    }
    v8f c = {};
    c = __builtin_amdgcn_wmma_f32_16x16x4_f32(false, a[0], false, b[0], (short)0, c, false, false);
    c = __builtin_amdgcn_wmma_f32_16x16x4_f32(false, a[1], false, b[1], (short)0, c, false, false);
    c = __builtin_amdgcn_wmma_f32_16x16x4_f32(false, a[2], false, b[2], (short)0, c, false, false);
    c = __builtin_amdgcn_wmma_f32_16x16x4_f32(false, a[3], false, b[3], (short)0, c, false, false);
    __syncthreads();
    #pragma unroll
    for (int j = 0; j < 8; ++j)                 // C/D layout: lane halves = M 0-7 / 8-15
      P[(half * 8 + j) * ROWP + mn] = c[j];
    __syncthreads();
  }

  // Combine halves: total = P_hi x P_lo.  Copy P_hi (bottom-right 8x8) to TA top-left;
  // TA[m<8][k>=8] is still zero, so chunks q=0,1 alone give the top-left product.
  #pragma unroll
  for (int e0 = 0; e0 < 64; e0 += 32) {
    int e = e0 + lane;
    int r = e >> 3, c2 = e & 7;
    TA[r * ROWP + c2] = P[(8 + r) * ROWP + (8 + c2)];
  }
  __syncthreads();
  {
    v2f a[2], b[2];
    #pragma unroll
    for (int q = 0; q < 2; ++q) {
      int kk = 4 * q + 2 * half;
      a[q].x = TA[mn * ROWP + kk];
      a[q].y = TA[mn * ROWP + kk + 1];
      b[q].x = P[kk * ROWP + mn];
      b[q].y = P[(kk + 1) * ROWP + mn];
    }
    v8f c = {};
    c = __builtin_amdgcn_wmma_f32_16x16x4_f32(false, a[0], false, b[0], (short)0, c, false, false);
    c = __builtin_amdgcn_wmma_f32_16x16x4_f32(false, a[1], false, b[1], (short)0, c, false, false);
    __syncthreads();
    #pragma unroll
    for (int j = 0; j < 8; ++j)
      P[(half * 8 + j) * ROWP + mn] = c[j];
  }
  __syncthreads();

  // Epilogue in fp64 on lane 0:  U = X + iY, X = P[0:4][0:4], Y = P[4:8][0:4].
  if (lane == 0) {
    double U00r = P[0*ROWP+0], U00i = P[4*ROWP+0];
    double U01r = P[0*ROWP+1], U01i = P[4*ROWP+1];
    double U22r = P[2*ROWP+2], U22i = P[6*ROWP+2];
    double U23r = P[2*ROWP+3], U23i = P[6*ROWP+3];
    double p0r = psi[0], p0i = psi[1];
    double p1r = psi[2], p1i = psi[3];
    double p2r = psi[4], p2i = psi[5];
    double p3r = psi[6], p3i = psi[7];
    // psi_f[0], psi_f[2] (U is block-diagonal)
    double f0r = U00r*p0r - U00i*p0i + U01r*p1r - U01i*p1i;
    double f0i = U00r*p0i + U00i*p0r + U01r*p1i + U01i*p1r;
    double f2r = U22r*p2r - U22i*p2i + U23r*p3r - U23i*p3i;
    double f2i = U22r*p2i + U22i*p2r + U23r*p3i + U23i*p3r;
    double th = theta[0];
    double st, ct;   sincos(th, &st, &ct);
    double s2t, c2t; sincos(2.0 * th, &s2t, &c2t);
    // a01 = e^{-i th} f0
    double a01r = ct*f0r + st*f0i;
    double a01i = ct*f0i - st*f0r;
    // a11 = e^{-i(2th+pi)} f2 = -e^{-2i th} f2
    double a11r = -(c2t*f2r + s2t*f2i);
    double a11i = -(c2t*f2i - s2t*f2r);
    double zr = 1.0 + 2.0*a01r + a11r;
    double zi =       2.0*a01i + a11i;
    double F = 0.05 * (zr*zr + zi*zi + 1.0
                       + 2.0*(a01r*a01r + a01i*a01i)
                       + (a11r*a11r + a11i*a11i));
    F = F < 0.0 ? 0.0 : (F > 1.0 ? 1.0 : F);
    out[0] = (float)F;
  }
}

extern "C" void kernel_launch(void* const* d_in, const int* in_sizes, int n_in,
                              void* d_out, int out_size, void* d_ws, size_t ws_size,
                              hipStream_t stream) {
  (void)in_sizes; (void)n_in; (void)out_size; (void)d_ws; (void)ws_size;
  const double* theta = (const double*)d_in[0];   // [1]  float64
  const double* phi   = (const double*)d_in[1];   // [99] float64
  const double* psi   = (const double*)d_in[2];   // [4]  complex128 -> 8 doubles
  const double* T     = (const double*)d_in[3];   // [1]  float64
  float* out = (float*)d_out;                     // [1]
  qubits2_bell_kernel<<<1, 32, 0, stream>>>(theta, phi, psi, T, out);
}